// Embed3D_20280835572091
// MI455X (gfx1250) — compile-verified
//
#include <hip/hip_runtime.h>

typedef __attribute__((ext_vector_type(4))) float v4f;
typedef __attribute__((ext_vector_type(2))) float v2f;

#define NFREQ 20          // div_term length (D=40, stride 2)
#define CHUNKS_PER_ROW 30 // 120 f32 outputs / 4 per thread

#if defined(__AMDGCN__) && __has_builtin(__builtin_amdgcn_global_load_async_to_lds_b32)
#define EMB_ASYNC_LDS 1
#else
#define EMB_ASYNC_LDS 0
#endif

__global__ __launch_bounds__(256) void Embed3D_kernel(
    const float* __restrict__ x,        // (N*L, 4) flat
    const float* __restrict__ div_term, // (20,)
    float* __restrict__ out,            // (N*L, 120) flat
    unsigned int n_chunks)              // N*L*30
{
    // Stage the 80-byte frequency table into LDS once per block.
    __shared__ __align__(16) float s_dt[32];

    if (threadIdx.x < NFREQ) {
#if EMB_ASYNC_LDS
        // gfx1250 async copy: global -> LDS, tracked by ASYNCcnt.
        // Builtin signature (from clang diagnostic): (AS1 int*, AS3 int*, Ii, Ii)
        __builtin_amdgcn_global_load_async_to_lds_b32(
            (__attribute__((address_space(1))) int*)(void*)(div_term + threadIdx.x),
            (__attribute__((address_space(3))) int*)(&s_dt[threadIdx.x]),
            0, 0);
#else
        s_dt[threadIdx.x] = div_term[threadIdx.x];
#endif
    }
#if EMB_ASYNC_LDS
    asm volatile("s_wait_asynccnt 0" ::: "memory");
#endif
    __syncthreads();

    unsigned int g = blockIdx.x * blockDim.x + threadIdx.x;
    if (g >= n_chunks) return;

    // g -> (row, channel c in 0..2, even freq index i0)
    unsigned int row = g / CHUNKS_PER_ROW;
    unsigned int r   = g - row * CHUNKS_PER_ROW;   // 0..29
    unsigned int c   = r / 10u;                    // 0..2
    unsigned int i0  = (r - c * 10u) * 2u;         // 0,2,..,18

    // One 4-byte global load; 10 adjacent lanes share the same address.
    float p = x[row * 4u + 1u + c];

    // 8-byte LDS load (i0 even, s_dt 16B aligned).
    v2f dt = *reinterpret_cast<const v2f*>(&s_dt[i0]);

    float a0 = p * dt.x;
    float a1 = p * dt.y;

    // Args are in [0,1): hardware v_sin_f32/v_cos_f32 are accurate here,
    // no range reduction required.
    v4f o;
    o.x = __sinf(a0);
    o.y = __cosf(a0);
    o.z = __sinf(a1);
    o.w = __cosf(a1);

    // Streaming 128-bit store: 252 MB output > 192 MB L2, use NT hint.
    __builtin_nontemporal_store(o, reinterpret_cast<v4f*>(out) + g);
}

extern "C" void kernel_launch(void* const* d_in, const int* in_sizes, int n_in,
                              void* d_out, int out_size, void* d_ws, size_t ws_size,
                              hipStream_t stream) {
    const float* x        = (const float*)d_in[0];
    const float* div_term = (const float*)d_in[1];
    float* out            = (float*)d_out;

    unsigned int n_chunks = (unsigned int)(out_size / 4);  // 4 floats per thread
    dim3 block(256);
    dim3 grid((n_chunks + block.x - 1) / block.x);
    Embed3D_kernel<<<grid, block, 0, stream>>>(x, div_term, out, n_chunks);
}